// Cell_5377299054721
// MI455X (gfx1250) — compile-verified
//
#include <hip/hip_runtime.h>

typedef __attribute__((ext_vector_type(2))) float v2f;
typedef __attribute__((ext_vector_type(4))) float v4f;
typedef __attribute__((ext_vector_type(8))) float v8f;

#define WAVES_PER_BLOCK 8

__device__ __forceinline__ float leakyf(float x) { return x >= 0.f ? x : 0.2f * x; }

// ---------------------------------------------------------------- zero fill
__global__ void k_zero(float* __restrict__ p, long n) {
    long i  = blockIdx.x * (long)blockDim.x + threadIdx.x;
    long st = (long)gridDim.x * blockDim.x;
    for (; i < n; i += st) p[i] = 0.f;
}

// ------------------------------------------------ edge scatter-sum (segment_sum)
// one wave per edge; each lane adds 2 consecutive floats with f32 atomics
__global__ void k_scatter_add(const float* __restrict__ X, int ldx,
                              const int* __restrict__ src, const int* __restrict__ dst,
                              float* __restrict__ out, int M) {
    long gid   = blockIdx.x * (long)blockDim.x + threadIdx.x;
    long total = (long)M * 32;
    if (gid >= total) return;
    int e = (int)(gid >> 5);
    int l = (int)(gid & 31);
    int s = src[e], d = dst[e];
    const float* xp = X + (size_t)s * ldx + l * 2;
    float*       op = out + (size_t)d * 64 + l * 2;
    atomicAdd(op,     xp[0]);
    atomicAdd(op + 1, xp[1]);
}

// ------------------------------------------------ generic node GEMM (wave-level WMMA)
// out[16x64 tile] = X(16xK, row stride ldx) @ W(Kx64) + bias [+ out if accumulate]
// W staged in LDS pair-packed: sWp[k2*128 + col*2 + j] = W[(2*k2+j)*64 + col]
// -> B fragment is one aligned ds_load_b64.
__global__ void k_gemm_node(const float* __restrict__ X, int ldx, int K,
                            const float* __restrict__ W, const float* __restrict__ bias,
                            float* __restrict__ out, int ldo, int nrows, int accumulate) {
    __shared__ float sWp[96 * 128];                  // up to K=192
    for (int i = threadIdx.x; i < (K >> 1) * 64; i += blockDim.x) {
        int k2 = i >> 6, col = i & 63;
        sWp[k2 * 128 + col * 2 + 0] = W[(size_t)(2 * k2 + 0) * 64 + col];
        sWp[k2 * 128 + col * 2 + 1] = W[(size_t)(2 * k2 + 1) * 64 + col];
    }
    __syncthreads();

    int lane = threadIdx.x & 31;
    int wib  = threadIdx.x >> 5;
    int tile = blockIdx.x * WAVES_PER_BLOCK + wib;
    int row0 = tile * 16;
    if (row0 >= nrows) return;           // wave-uniform: EXEC stays all-ones for WMMA
    int half = lane >> 4;                // 0: K pair {0,1}; 1: K pair {2,3}
    int r    = lane & 15;
    const float* xrow = X + (size_t)(row0 + r) * ldx;

    v8f z = {0.f,0.f,0.f,0.f,0.f,0.f,0.f,0.f};
    v8f acc[4] = {z, z, z, z};

    for (int k = 0; k < K; k += 4) {
        int c0 = k + 2 * half;           // even
        v2f a;
        a.x = xrow[c0];
        a.y = xrow[c0 + 1];
#pragma unroll
        for (int t = 0; t < 4; ++t) {
            int col = t * 16 + r;
            v2f b = *(const v2f*)&sWp[(c0 >> 1) * 128 + col * 2];
            acc[t] = __builtin_amdgcn_wmma_f32_16x16x4_f32(
                false, a, false, b, (short)0, acc[t], false, false);
        }
    }
#pragma unroll
    for (int t = 0; t < 4; ++t) {
        int col  = t * 16 + r;
        float bv = bias[col];
#pragma unroll
        for (int i = 0; i < 8; ++i) {
            int row = row0 + 8 * half + i;     // C/D layout: VGPR i -> M=i (+8 for hi lanes)
            float v = acc[t][i] + bv;
            float* o = out + (size_t)row * ldo + col;
            if (accumulate) v += *o;
            *o = v;
        }
    }
}

// ------------------------------------------------ edge GEMM: 800k x 192 @ 192 x 64
// A rows gathered inline from [V_new[src] | leaky(E) | V_new[dst]] (V_new is L2-resident).
// W_S staged into LDS via gfx1250 async-to-LDS (ASYNCcnt path), then pair-packed
// so B fragments are single ds_load_b64.
__global__ void k_edge_gemm(const float* __restrict__ Vn, const float* __restrict__ E,
                            const int* __restrict__ src, const int* __restrict__ dst,
                            const float* __restrict__ WS, const float* __restrict__ bS,
                            float* __restrict__ Eout, int M) {
    __shared__ float sRaw[192 * 64];     // raw W_S (48KB)
    __shared__ float sWp[96 * 128];      // pair-packed W_S (48KB); WGP has 320KB LDS

    // async fill of raw W_S: 16B per issue, no VGPR staging, tracked by ASYNCcnt
    for (int i = threadIdx.x * 4; i < 192 * 64; i += blockDim.x * 4) {
        unsigned     lds_off = (unsigned)(size_t)&sRaw[i];   // low 32 bits = LDS offset
        const float* gp      = WS + i;
        asm volatile("global_load_async_to_lds_b128 %0, %1, off"
                     :: "v"(lds_off), "v"(gp) : "memory");
    }
    asm volatile("s_wait_asynccnt 0x0" ::: "memory");
    __syncthreads();

    // repack to paired layout for b64 B loads
    for (int i = threadIdx.x; i < 96 * 64; i += blockDim.x) {
        int k2 = i >> 6, col = i & 63;
        sWp[k2 * 128 + col * 2 + 0] = sRaw[(2 * k2 + 0) * 64 + col];
        sWp[k2 * 128 + col * 2 + 1] = sRaw[(2 * k2 + 1) * 64 + col];
    }
    __syncthreads();

    int lane = threadIdx.x & 31;
    int wib  = threadIdx.x >> 5;
    int tile = blockIdx.x * WAVES_PER_BLOCK + wib;
    int e0   = tile * 16;
    if (e0 >= M) return;                 // wave-uniform
    int half = lane >> 4;
    int r    = lane & 15;
    int e    = e0 + r;
    const float* pS = Vn + (size_t)src[e] * 64;
    const float* pE = E  + (size_t)e      * 64;
    const float* pD = Vn + (size_t)dst[e] * 64;

    v8f z = {0.f,0.f,0.f,0.f,0.f,0.f,0.f,0.f};
    v8f acc[4] = {z, z, z, z};

    for (int k = 0; k < 192; k += 4) {
        int c0 = k + 2 * half;           // segment choice wave-uniform (4-aligned steps)
        v2f a;
        if (c0 < 64)       { a.x = pS[c0];              a.y = pS[c0 + 1]; }
        else if (c0 < 128) { a.x = leakyf(pE[c0 - 64]); a.y = leakyf(pE[c0 - 63]); }
        else               { a.x = pD[c0 - 128];        a.y = pD[c0 - 127]; }
#pragma unroll
        for (int t = 0; t < 4; ++t) {
            int col = t * 16 + r;
            v2f b = *(const v2f*)&sWp[(c0 >> 1) * 128 + col * 2];
            acc[t] = __builtin_amdgcn_wmma_f32_16x16x4_f32(
                false, a, false, b, (short)0, acc[t], false, false);
        }
    }
#pragma unroll
    for (int t = 0; t < 4; ++t) {
        int col  = t * 16 + r;
        float bv = bS[col];
#pragma unroll
        for (int i = 0; i < 8; ++i)
            Eout[(size_t)(e0 + 8 * half + i) * 64 + col] = acc[t][i] + bv;
    }
}

// ------------------------------------------------ per-column sum / sum-of-squares
__global__ void k_col_stats(const float* __restrict__ X, long rows,
                            float* __restrict__ s, float* __restrict__ q) {
    int  c   = threadIdx.x & 63;
    int  grp = threadIdx.x >> 6;                 // 4 row-streams per 256-thread block
    long r   = (long)blockIdx.x * 4 + grp;
    long st  = (long)gridDim.x * 4;
    float sa = 0.f, qa = 0.f;
    for (; r < rows; r += st) {
        float v = X[r * 64 + c];
        sa += v; qa += v * v;
    }
    atomicAdd(&s[c], sa);
    atomicAdd(&q[c], qa);
}

// ------------------------------------------------ BN (train mode) + leaky + residual
// float4 path: this is the memory-bound tail, so use b128 loads/stores
__global__ void k_bn_apply(const float* __restrict__ Xp, const float* __restrict__ R,
                           const float* __restrict__ s, const float* __restrict__ q,
                           float inv_cnt, const float* __restrict__ g,
                           const float* __restrict__ b, float* __restrict__ out, long rows) {
    long i     = blockIdx.x * (long)blockDim.x + threadIdx.x;
    long total = rows * 16;                       // float4 groups per row = 16
    long st    = (long)gridDim.x * blockDim.x;
    const v4f* X4 = (const v4f*)Xp;
    const v4f* R4 = (const v4f*)R;
    v4f*       O4 = (v4f*)out;
    for (; i < total; i += st) {
        int c0 = (int)(i & 15) * 4;
        v4f x = X4[i], rr = R4[i], y;
#pragma unroll
        for (int j = 0; j < 4; ++j) {
            int   c   = c0 + j;
            float mu  = s[c] * inv_cnt;
            float var = q[c] * inv_cnt - mu * mu; // biased, like jnp.var
            float xh  = (x[j] - mu) * rsqrtf(var + 1e-5f);
            float t   = g[c] * xh + b[c];
            y[j]      = leakyf(t) + rr[j];
        }
        O4[i] = y;
    }
}

extern "C" void kernel_launch(void* const* d_in, const int* in_sizes, int n_in,
                              void* d_out, int out_size, void* d_ws, size_t ws_size,
                              hipStream_t stream) {
    const float* V       = (const float*)d_in[0];
    const float* E       = (const float*)d_in[1];
    const int*   src     = (const int*)d_in[2];
    const int*   dst     = (const int*)d_in[3];
    const float* W_ops   = (const float*)d_in[4];   // 5 x 64 x 64
    const float* b_ops   = (const float*)d_in[5];   // 5 x 64
    const float* W_cat   = (const float*)d_in[6];   // 192 x 64
    const float* b_cat   = (const float*)d_in[7];
    const float* W_S     = (const float*)d_in[8];   // 192 x 64
    const float* b_S     = (const float*)d_in[9];
    const float* gamma_V = (const float*)d_in[10];
    const float* beta_V  = (const float*)d_in[11];
    const float* gamma_E = (const float*)d_in[12];
    const float* beta_E  = (const float*)d_in[13];

    const int D = 64;
    const int N = in_sizes[0] / D;
    const int M = in_sizes[2];

    // workspace layout (floats): AGG(N*64) | S(N*192: s1|s2|s3) | Vn(N*64) | Epre(M*64) | stats(256)
    float* ws    = (float*)d_ws;
    float* AGG   = ws;
    float* S     = AGG  + (size_t)N * 64;
    float* Vn    = S    + (size_t)N * 192;
    float* Epre  = Vn   + (size_t)N * 64;
    float* stats = Epre + (size_t)M * 64;
    float* sumV = stats, *sqV = stats + 64, *sumE = stats + 128, *sqE = stats + 192;

    float* outV = (float*)d_out;
    float* outE = outV + (size_t)N * 64;

    const int ntiles = (N + 15) / 16, mtiles = (M + 15) / 16;
    const int gN = (ntiles + WAVES_PER_BLOCK - 1) / WAVES_PER_BLOCK;
    const int gM = (mtiles + WAVES_PER_BLOCK - 1) / WAVES_PER_BLOCK;
    const int gScat = (int)(((long)M * 32 + 255) / 256);
    dim3 blk(256);

    // AGG_V = segment_sum(V[src], dst)
    k_zero<<<1024, blk, 0, stream>>>(AGG, (long)N * 64);
    k_zero<<<1, blk, 0, stream>>>(stats, 256);
    k_scatter_add<<<gScat, blk, 0, stream>>>(V, 64, src, dst, AGG, M);

    // s1 = AGG @ W0 + b0 ; s2 = AGG @ W1 + b1 + s1 @ W2 + b2
    k_gemm_node<<<gN, blk, 0, stream>>>(AGG, 64, 64, W_ops,            b_ops,       S,       192, N, 0);
    k_gemm_node<<<gN, blk, 0, stream>>>(AGG, 64, 64, W_ops + 4096,     b_ops + 64,  S + 64,  192, N, 0);
    k_gemm_node<<<gN, blk, 0, stream>>>(S,   192, 64, W_ops + 2*4096,  b_ops + 128, S + 64,  192, N, 1);

    // AGG_s1 = segment_sum(s1[src], dst) ; s3 = AGG_s1 @ W3 + b3 + s2 @ W4 + b4
    k_zero<<<1024, blk, 0, stream>>>(AGG, (long)N * 64);
    k_scatter_add<<<gScat, blk, 0, stream>>>(S, 192, src, dst, AGG, M);
    k_gemm_node<<<gN, blk, 0, stream>>>(AGG,    64, 64, W_ops + 3*4096, b_ops + 192, S + 128, 192, N, 0);
    k_gemm_node<<<gN, blk, 0, stream>>>(S + 64, 192, 64, W_ops + 4*4096, b_ops + 256, S + 128, 192, N, 1);

    // V_new = [s1|s2|s3] @ W_cat + b_cat   (concat is free: S has row stride 192)
    k_gemm_node<<<gN, blk, 0, stream>>>(S, 192, 192, W_cat, b_cat, Vn, 64, N, 0);

    // E_pre = [V_new[src] | leaky(E) | V_new[dst]] @ W_S + b_S
    k_edge_gemm<<<gM, blk, 0, stream>>>(Vn, E, src, dst, W_S, b_S, Epre, M);

    // BN stats + apply (BN + leaky + residual)
    k_col_stats<<<512,  blk, 0, stream>>>(Vn,   (long)N, sumV, sqV);
    k_col_stats<<<2048, blk, 0, stream>>>(Epre, (long)M, sumE, sqE);
    k_bn_apply<<<2048, blk, 0, stream>>>(Vn,   V, sumV, sqV, 1.f / (float)N, gamma_V, beta_V, outV, (long)N);
    k_bn_apply<<<4096, blk, 0, stream>>>(Epre, E, sumE, sqE, 1.f / (float)M, gamma_E, beta_E, outE, (long)M);
}